// SAGPoolNet_29892972380782
// MI455X (gfx1250) — compile-verified
//
#include <hip/hip_runtime.h>
#include <math.h>

typedef __attribute__((ext_vector_type(2))) float v2f;
typedef __attribute__((ext_vector_type(8))) float v8f;

#define NP   128          // nodes per graph
#define HD   64           // hidden / input feature dim
#define EPG  2048         // edges per graph
#define ETOT (512 * EPG)  // total edges
#define KSEL 64           // top-k
#define LDA  68           // padded stride for 64-wide LDS buffers
#define LDH  212          // padded stride for 192-wide concat buffer

// ---- LDS layout (in floats) ----
#define OFF_XB   0                        // 128*68
#define OFF_HW   (OFF_XB + NP * LDA)      // 128*68 GEMM output scratch
#define OFF_WT   (OFF_HW + NP * LDA)      // 64*68  transposed weight
#define OFF_HCAT (OFF_WT + HD * LDA)      // 128*212 concat(h1,h2,h3)
#define OFF_EDG  (OFF_HCAT + NP * LDH)    // 2048 packed (src | dst<<8) ints
#define OFF_WN   (OFF_EDG + EPG)          // 2048 edge norms
#define OFF_DEG  (OFF_WN + EPG)           // 128
#define OFF_DINV (OFF_DEG + NP)           // 128
#define OFF_SLIN (OFF_DINV + NP)          // 128
#define OFF_SC   (OFF_SLIN + NP)          // 128
#define OFF_TANH (OFF_SC + NP)            // 128
#define OFF_SEL  (OFF_TANH + NP)          // 128
#define OFF_POOL (OFF_SEL + NP)           // 384
#define OFF_Z1   (OFF_POOL + 384)         // 64
#define OFF_Z2   (OFF_Z1 + 64)            // 32
#define OFF_Z3   (OFF_Z2 + 32)            // 4 (+pad)
#define LDS_FLOATS (OFF_Z3 + 8)
#define SMEM_BYTES (LDS_FLOATS * 4)

// 128x64 = A(128x64, stride lda) @ Wt(64 cols stored transposed, stride LDA)
// using V_WMMA_F32_16X16X4_F32. 8 waves: wave w owns M-strip [16w,16w+16),
// holds 4 N-tile accumulators, 16 K-steps of 4.
__device__ __forceinline__ void gemm_wmma(const float* __restrict__ A, int lda,
                                          const float* __restrict__ WT,
                                          float* __restrict__ HW, int tid) {
  const int wave = tid >> 5;
  const int lane = tid & 31;
  const int half = lane >> 4;   // 0: K pair {0,1}, 1: K pair {2,3}
  const int lm   = lane & 15;
  const float* arow = A + (wave * 16 + lm) * lda;
  const float* w0 = WT + ( 0 + lm) * LDA;
  const float* w1 = WT + (16 + lm) * LDA;
  const float* w2 = WT + (32 + lm) * LDA;
  const float* w3 = WT + (48 + lm) * LDA;
  v8f c0 = {}, c1 = {}, c2 = {}, c3 = {};
#pragma unroll
  for (int kk = 0; kk < 16; ++kk) {
    const int k0 = kk * 4 + half * 2;
    v2f a  = *(const v2f*)(arow + k0);
    v2f b0 = *(const v2f*)(w0 + k0);
    v2f b1 = *(const v2f*)(w1 + k0);
    v2f b2 = *(const v2f*)(w2 + k0);
    v2f b3 = *(const v2f*)(w3 + k0);
    c0 = __builtin_amdgcn_wmma_f32_16x16x4_f32(false, a, false, b0, (short)0, c0, false, false);
    c1 = __builtin_amdgcn_wmma_f32_16x16x4_f32(false, a, false, b1, (short)0, c1, false, false);
    c2 = __builtin_amdgcn_wmma_f32_16x16x4_f32(false, a, false, b2, (short)0, c2, false, false);
    c3 = __builtin_amdgcn_wmma_f32_16x16x4_f32(false, a, false, b3, (short)0, c3, false, false);
  }
#pragma unroll
  for (int v = 0; v < 8; ++v) {
    const int m = wave * 16 + v + half * 8;
    HW[m * LDA +  0 + lm] = c0[v];
    HW[m * LDA + 16 + lm] = c1[v];
    HW[m * LDA + 32 + lm] = c2[v];
    HW[m * LDA + 48 + lm] = c3[v];
  }
}

__global__ void __launch_bounds__(256) sagpool_fused(
    const float* __restrict__ x, const int* __restrict__ ei,
    const float* __restrict__ W1, const float* __restrict__ b1,
    const float* __restrict__ W2, const float* __restrict__ b2,
    const float* __restrict__ W3, const float* __restrict__ b3,
    const float* __restrict__ Wsc, const float* __restrict__ bsc,
    const float* __restrict__ Wg, const float* __restrict__ bg,
    const float* __restrict__ Wl2, const float* __restrict__ bl2,
    const float* __restrict__ Wl3, const float* __restrict__ bl3,
    float* __restrict__ out) {
  extern __shared__ __align__(16) float smem[];
  const int g = blockIdx.x;
  const int tid = threadIdx.x;

  float* XB   = smem + OFF_XB;
  float* HW   = smem + OFF_HW;
  float* WT   = smem + OFF_WT;
  float* HCAT = smem + OFF_HCAT;
  int*   EDG  = (int*)(smem + OFF_EDG);
  float* WN   = smem + OFF_WN;
  float* DEG  = smem + OFF_DEG;
  float* DINV = smem + OFF_DINV;
  float* SLIN = smem + OFF_SLIN;
  float* SC   = smem + OFF_SC;
  float* TANH = smem + OFF_TANH;
  float* SEL  = smem + OFF_SEL;
  float* POOL = smem + OFF_POOL;
  float* Z1   = smem + OFF_Z1;
  float* Z2   = smem + OFF_Z2;
  float* Z3   = smem + OFF_Z3;

  // ---- stage in node features + packed local edges; zero degree ----
  for (int i = tid; i < NP * HD; i += 256) {
    const int m = i >> 6, c = i & 63;
    XB[m * LDA + c] = x[((size_t)g * NP + m) * HD + c];
  }
  for (int e = tid; e < EPG; e += 256) {
    const int sl = ei[(size_t)g * EPG + e] & (NP - 1);
    const int dl = ei[(size_t)ETOT + (size_t)g * EPG + e] & (NP - 1);
    EDG[e] = sl | (dl << 8);
  }
  if (tid < NP) DEG[tid] = 0.0f;
  __syncthreads();

  // ---- degree (dst counts), dinv, per-edge norm ----
  for (int e = tid; e < EPG; e += 256)
    atomicAdd(&DEG[EDG[e] >> 8], 1.0f);
  __syncthreads();
  if (tid < NP) DINV[tid] = rsqrtf(DEG[tid] + 1.0f);
  __syncthreads();
  for (int e = tid; e < EPG; e += 256) {
    const int p = EDG[e];
    WN[e] = DINV[p & 255] * DINV[p >> 8];
  }
  __syncthreads();

  // ---- 3 GCN layers, outputs into HCAT columns [64L, 64L+64) ----
  const float* Wl[3] = {W1, W2, W3};
  const float* bl[3] = {b1, b2, b3};
  for (int L = 0; L < 3; ++L) {
    // stage W transposed: WT[o][k] = W[k][o]
    for (int i = tid; i < HD * HD; i += 256)
      WT[(i & 63) * LDA + (i >> 6)] = Wl[L][i];
    __syncthreads();

    const float* Ain = (L == 0) ? XB : (HCAT + (L - 1) * HD);
    const int lda = (L == 0) ? LDA : LDH;
    gemm_wmma(Ain, lda, WT, HW, tid);  // HW = h_in @ W
    __syncthreads();

    // init with self-loop term + bias
    float* Hout = HCAT + L * HD;
    for (int i = tid; i < NP * HD; i += 256) {
      const int m = i >> 6, c = i & 63;
      Hout[m * LDH + c] = DINV[m] * DINV[m] * HW[m * LDA + c] + bl[L][c];
    }
    __syncthreads();

    // edge scatter: each 64-thread group (two waves) handles all 64 columns of one edge
    {
      const int col = tid & 63;
      for (int e = tid >> 6; e < EPG; e += 4) {
        const int p = EDG[e];
        atomicAdd(&Hout[(p >> 8) * LDH + col], WN[e] * HW[(p & 255) * LDA + col]);
      }
    }
    __syncthreads();

    // relu in place
    for (int i = tid; i < NP * HD; i += 256) {
      const int m = i >> 6, c = i & 63;
      Hout[m * LDH + c] = fmaxf(Hout[m * LDH + c], 0.0f);
    }
    __syncthreads();
  }

  // ---- SAGPool score GCN: slin = h @ Ws (192 -> 1) ----
  if (tid < NP) {
    float acc = 0.0f;
    const float* hr = HCAT + tid * LDH;
    for (int j = 0; j < 3 * HD; ++j) acc += hr[j] * Wsc[j];
    SLIN[tid] = acc;
  }
  __syncthreads();
  if (tid < NP) SC[tid] = DINV[tid] * DINV[tid] * SLIN[tid] + bsc[0];
  __syncthreads();
  for (int e = tid; e < EPG; e += 256) {
    const int p = EDG[e];
    atomicAdd(&SC[p >> 8], WN[e] * SLIN[p & 255]);
  }
  __syncthreads();

  // ---- top-64 by rank counting (tie-break: lower index wins) ----
  if (tid < NP) {
    const float s = SC[tid];
    int rank = 0;
    for (int m = 0; m < NP; ++m) {
      const float sm = SC[m];
      rank += (sm > s) || (sm == s && m < tid);
    }
    SEL[tid]  = (rank < KSEL) ? 1.0f : 0.0f;
    TANH[tid] = tanhf(s);
  }
  __syncthreads();

  // ---- pooled = [max ; mean] over selected, scaled by tanh(score) ----
  if (tid < 3 * HD) {
    float mx = -INFINITY, sum = 0.0f;
    for (int m = 0; m < NP; ++m) {
      if (SEL[m] > 0.5f) {
        const float v = HCAT[m * LDH + tid] * TANH[m];
        mx = fmaxf(mx, v);
        sum += v;
      }
    }
    POOL[tid] = mx;
    POOL[3 * HD + tid] = sum * (1.0f / KSEL);
  }
  __syncthreads();

  // ---- MLP head ----
  if (tid < 64) {
    float acc = bg[tid];
    for (int j = 0; j < 6 * HD; ++j) acc += POOL[j] * Wg[j * 64 + tid];
    Z1[tid] = fmaxf(acc, 0.0f);
  }
  __syncthreads();
  if (tid < 32) {
    float acc = bl2[tid];
    for (int i = 0; i < 64; ++i) acc += Z1[i] * Wl2[i * 32 + tid];
    Z2[tid] = fmaxf(acc, 0.0f);
  }
  __syncthreads();
  if (tid < 4) {
    float acc = bl3[tid];
    for (int i = 0; i < 32; ++i) acc += Z2[i] * Wl3[i * 4 + tid];
    Z3[tid] = acc;
  }
  __syncthreads();
  if (tid < 4) {
    const float m0 = fmaxf(fmaxf(Z3[0], Z3[1]), fmaxf(Z3[2], Z3[3]));
    const float lse = m0 + logf(expf(Z3[0] - m0) + expf(Z3[1] - m0) +
                                expf(Z3[2] - m0) + expf(Z3[3] - m0));
    out[(size_t)g * 4 + tid] = Z3[tid] - lse;
  }
}

extern "C" void kernel_launch(void* const* d_in, const int* in_sizes, int n_in,
                              void* d_out, int out_size, void* d_ws, size_t ws_size,
                              hipStream_t stream) {
  (void)in_sizes; (void)n_in; (void)out_size; (void)d_ws; (void)ws_size;
  const float* x   = (const float*)d_in[0];
  const int*   ei  = (const int*)d_in[1];
  // d_in[2]=batch, d_in[3]=batch_size, d_in[4]=edge_attr : unused
  const float* W1  = (const float*)d_in[5];
  const float* b1  = (const float*)d_in[6];
  const float* W2  = (const float*)d_in[7];
  const float* b2  = (const float*)d_in[8];
  const float* W3  = (const float*)d_in[9];
  const float* b3  = (const float*)d_in[10];
  const float* Wsc = (const float*)d_in[11];
  const float* bsc = (const float*)d_in[12];
  const float* Wg  = (const float*)d_in[13];
  const float* bg  = (const float*)d_in[14];
  const float* Wl2 = (const float*)d_in[15];
  const float* bl2 = (const float*)d_in[16];
  const float* Wl3 = (const float*)d_in[17];
  const float* bl3 = (const float*)d_in[18];
  float* out = (float*)d_out;

  hipFuncSetAttribute((const void*)sagpool_fused,
                      hipFuncAttributeMaxDynamicSharedMemorySize, SMEM_BYTES);
  sagpool_fused<<<512, 256, SMEM_BYTES, stream>>>(
      x, ei, W1, b1, W2, b2, W3, b3, Wsc, bsc, Wg, bg, Wl2, bl2, Wl3, bl3, out);
}